// MDN_6485400617753
// MI455X (gfx1250) — compile-verified
//
#include <hip/hip_runtime.h>

typedef __attribute__((ext_vector_type(2))) float v2f;
typedef __attribute__((ext_vector_type(4))) float v4f;
typedef __attribute__((ext_vector_type(8))) float v8f;

#define B_TOT 65536
#define CONDN 32
#define HID   256
#define KMIX  16
#define DDIM  16

#define ZSTRIDE 260            // 256 + 4 pad -> conflict-free LDS stores & b64 A-frag loads
#define WAVES_PER_BLOCK 2

#define WMMA_F32(A, Bv, C) \
  __builtin_amdgcn_wmma_f32_16x16x4_f32(false, (A), false, (Bv), (short)0, (C), false, false)

// branch-free tanh: e = exp(-2|x|) in (0,1] (never overflows), tanh = sign(x)*(1-e)/(1+e)
static __device__ __forceinline__ float fast_tanh(float x) {
  const float e = __expf(-2.0f * __builtin_fabsf(x));
  const float r = (1.0f - e) * __builtin_amdgcn_rcpf(1.0f + e);
  return __builtin_copysignf(r, x);
}

// z (16x256, in LDS) @ W (256 x N) + bias, one 16-wide N-tile, f32 WMMA 16x16x4.
template<int N>
static __device__ __forceinline__
v8f gemm_z256(const float* __restrict__ W, const float* __restrict__ bias,
              const float* zrow, int nt, int l16, int half)
{
  const float bv = bias[nt * 16 + l16];
  v8f acc = {bv, bv, bv, bv, bv, bv, bv, bv};
  const float* wc = W + (size_t)(2 * half) * N + nt * 16 + l16;
#pragma unroll
  for (int kb = 0; kb < 64; ++kb) {
    v2f a = *(const v2f*)(zrow + 4 * kb);           // ds_load_b64, conflict-free
    v2f b;
    b.x = wc[(4 * kb + 0) * N];                     // weights: L2/WGP$ resident
    b.y = wc[(4 * kb + 1) * N];
    acc = WMMA_F32(a, b, acc);
  }
  return acc;
}

__global__ __launch_bounds__(WAVES_PER_BLOCK * 32)
void mdn_fused(const float* __restrict__ x,
               const float* __restrict__ W1,  const float* __restrict__ b1,
               const float* __restrict__ Wpi, const float* __restrict__ bpi,
               const float* __restrict__ Wsig, const float* __restrict__ bsig,
               const float* __restrict__ Wmu, const float* __restrict__ bmu,
               float* __restrict__ pi_out, float* __restrict__ sig_out,
               float* __restrict__ mu_out)
{
  __shared__ float zsh[WAVES_PER_BLOCK * 16 * ZSTRIDE];

  const int lane = threadIdx.x & 31;   // wave32 on gfx1250
  const int wave = threadIdx.x >> 5;
  const int l16  = lane & 15;
  const int half = lane >> 4;

  const long long row0 = ((long long)blockIdx.x * WAVES_PER_BLOCK + wave) * 16;
  float* zw = &zsh[wave * 16 * ZSTRIDE];

  // ---- x tile (16x32) as WMMA A fragments: lane holds row l16, K pair k0+2*half ----
  v2f ax[8];
  const float* xrow = x + (row0 + l16) * CONDN + 2 * half;
#pragma unroll
  for (int kb = 0; kb < 8; ++kb)
    ax[kb] = *(const v2f*)(xrow + 4 * kb);

  // ---- layer 1: z = tanh(x @ W1 + b1) -> LDS (C layout -> row-major with pad) ----
  for (int nt = 0; nt < HID / 16; ++nt) {
    const float bv = b1[nt * 16 + l16];
    v8f acc = {bv, bv, bv, bv, bv, bv, bv, bv};
    const float* wc = W1 + (size_t)(2 * half) * HID + nt * 16 + l16;
#pragma unroll
    for (int kb = 0; kb < 8; ++kb) {
      v2f b;
      b.x = wc[(4 * kb + 0) * HID];
      b.y = wc[(4 * kb + 1) * HID];
      acc = WMMA_F32(ax[kb], b, acc);
    }
#pragma unroll
    for (int v = 0; v < 8; ++v)
      zw[(v + 8 * half) * ZSTRIDE + nt * 16 + l16] = fast_tanh(acc[v]);
  }

  __syncthreads();

  const float* zrow = zw + l16 * ZSTRIDE + 2 * half;

  // ---- pi head: softmax(z @ Wpi + bpi); row's 16 logits live in one 16-lane half ----
  {
    v8f p = gemm_z256<KMIX>(Wpi, bpi, zrow, 0, l16, half);
#pragma unroll
    for (int v = 0; v < 8; ++v) {
      float t = p[v];
#pragma unroll
      for (int m = 1; m <= 8; m <<= 1)           // masks 1,2,4,8 stay inside half-wave
        t = fmaxf(t, __shfl_xor(t, m, 32));
      const float e = __expf(p[v] - t);
      float s = e;
#pragma unroll
      for (int m = 1; m <= 8; m <<= 1)
        s += __shfl_xor(s, m, 32);
      const long long row = row0 + v + 8 * half;
      __builtin_nontemporal_store(e * __builtin_amdgcn_rcpf(s),
                                  pi_out + row * KMIX + l16);
    }
  }

  // ---- fused sigma + mu heads: one z A-fragment feeds two WMMAs per k-block ----
  for (int nt = 0; nt < KMIX; ++nt) {            // nt == mixture index k (D == 16)
    const float bs = bsig[nt * 16 + l16];
    const float bm = bmu[nt * 16 + l16];
    v8f sacc = {bs, bs, bs, bs, bs, bs, bs, bs};
    v8f macc = {bm, bm, bm, bm, bm, bm, bm, bm};
    const float* ws = Wsig + (size_t)(2 * half) * (KMIX * DDIM) + nt * 16 + l16;
    const float* wm = Wmu  + (size_t)(2 * half) * (KMIX * DDIM) + nt * 16 + l16;
#pragma unroll
    for (int kb = 0; kb < 64; ++kb) {
      v2f a = *(const v2f*)(zrow + 4 * kb);        // shared A fragment
      v2f b0;
      b0.x = ws[(4 * kb + 0) * (KMIX * DDIM)];
      b0.y = ws[(4 * kb + 1) * (KMIX * DDIM)];
      sacc = WMMA_F32(a, b0, sacc);
      v2f b1;
      b1.x = wm[(4 * kb + 0) * (KMIX * DDIM)];
      b1.y = wm[(4 * kb + 1) * (KMIX * DDIM)];
      macc = WMMA_F32(a, b1, macc);
    }
#pragma unroll
    for (int v = 0; v < 8; ++v) {
      const float sv = __expf(sacc[v]);
      const long long row = row0 + v + 8 * half;
      // sigma: diagonal-embedded 16x16 block, 4 x b128 NT stores per lane (row i = l16)
      float* dst = sig_out + ((row * KMIX + nt) * DDIM + l16) * DDIM;
#pragma unroll
      for (int c = 0; c < 4; ++c) {
        v4f o;
        o.x = (l16 == 4 * c + 0) ? sv : 0.0f;
        o.y = (l16 == 4 * c + 1) ? sv : 0.0f;
        o.z = (l16 == 4 * c + 2) ? sv : 0.0f;
        o.w = (l16 == 4 * c + 3) ? sv : 0.0f;
        __builtin_nontemporal_store(o, (v4f*)(dst + 4 * c));
      }
      // mu
      __builtin_nontemporal_store(macc[v], mu_out + (row * KMIX + nt) * DDIM + l16);
    }
  }
}

extern "C" void kernel_launch(void* const* d_in, const int* in_sizes, int n_in,
                              void* d_out, int out_size, void* d_ws, size_t ws_size,
                              hipStream_t stream) {
  (void)in_sizes; (void)n_in; (void)out_size; (void)d_ws; (void)ws_size;
  const float* x    = (const float*)d_in[0];
  const float* W1   = (const float*)d_in[1];
  const float* b1   = (const float*)d_in[2];
  const float* Wpi  = (const float*)d_in[3];
  const float* bpi  = (const float*)d_in[4];
  const float* Wsig = (const float*)d_in[5];
  const float* bsig = (const float*)d_in[6];
  const float* Wmu  = (const float*)d_in[7];
  const float* bmu  = (const float*)d_in[8];

  float* pi_out  = (float*)d_out;                                   // B*K
  float* sig_out = pi_out + (size_t)B_TOT * KMIX;                   // B*K*D*D
  float* mu_out  = sig_out + (size_t)B_TOT * KMIX * DDIM * DDIM;    // B*K*D

  dim3 grid(B_TOT / (WAVES_PER_BLOCK * 16));   // 2048 blocks
  dim3 block(WAVES_PER_BLOCK * 32);            // 2 waves
  hipLaunchKernelGGL(mdn_fused, grid, block, 0, stream,
                     x, W1, b1, Wpi, bpi, Wsig, bsig, Wmu, bmu,
                     pi_out, sig_out, mu_out);
}